// ZebraFinchVocalTract_26328149524536
// MI455X (gfx1250) — compile-verified
//
#include <hip/hip_runtime.h>

// ---------------- model constants (double-precision fold -> f32, matching JAX trace) ----
namespace zf {
constexpr double dGAMMA  = 24000.0;
constexpr double dCH_INV = 45000000000.0;
constexpr double dLB_INV = 0.0001;
constexpr double dLG_INV = 1.0 / 82.0;
constexpr double dRB     = 5000000.0;
constexpr double dRH     = 600000.0;
constexpr double dDT     = 1.0 / (44100.0 * 2.0);

constexpr float GAMMA   = (float)dGAMMA;
constexpr float GG      = (float)(dGAMMA * dGAMMA);
constexpr float C1      = (float)(dLG_INV * dCH_INV);
constexpr float C2      = (float)(dRH * (dLB_INV + dLG_INV));
constexpr float C3      = (float)(dLG_INV * dCH_INV - dRB * dRH * dLB_INV * dLG_INV);
constexpr float C4      = (float)(dRH * dLB_INV * dLG_INV);
constexpr float C5      = (float)(dLB_INV / dLG_INV);
constexpr float C6      = (float)(dRB * dLB_INV);
constexpr float LG_INV  = (float)dLG_INV;
constexpr float LB_INV  = (float)dLB_INV;
constexpr float DT      = (float)dDT;
constexpr float HALF_DT = (float)(0.5 * dDT);
constexpr float DT6     = (float)(dDT / 6.0);
constexpr float INV_DT  = (float)(1.0 / dDT);
constexpr float T_IN    = 0.5f;
constexpr float R_REFL  = 0.1f;
constexpr float FACTOR  = 5.952507019042969f;

constexpr int HOP    = 256;
constexpr int SP     = 2;
constexpr int LSTEPS = HOP * SP;  // 512 steps per frame
constexpr int DELAY  = 8;         // TAU = 8, BUF_LEN = 9 -> pure 8-step FIFO
}  // namespace zf

// Per-frame invariants, hoisted out of the 512-step inner loop.
struct Frame {
    float ga;     // GG * a
    float b;      // beta
    float TA1;    // T_IN * A1
    float LGTA1;  // LG_INV * T_IN * A1
};

// dX/dt. k[0]=x1, k[2]=x3 are pass-throughs and handled by the caller.
// ldb = LG_INV * dbtau (stage-invariant); c4v = C4 * Vext.
__device__ __forceinline__ void zf_deriv(const float X[5], float c4v, const Frame& F,
                                         float ldb, float k[5]) {
    using namespace zf;
    const float x0 = X[0], x1 = X[1], x2 = X[2], x3 = X[3], x4 = X[4];
    const float x0sq = x0 * x0;
    // GG*(a + x0*(b + x0 - x0^2)) - GAMMA*(x0*x1)*(1 + x0)
    const float poly = fmaf(x0, F.b + x0 - x0sq, 0.0f);
    const float v01  = x0 * x1;
    const float k1   = fmaf(GG, poly, F.ga) - GAMMA * fmaf(x0, v01, v01);
    // k3 = -C1*x2 - C2*x3 + C3*x4 + LG_INV*(T_IN*A1*k1 + dbtau) + C4*Vext
    const float k3b  = fmaf(-C1, x2, fmaf(-C2, x3, fmaf(C3, x4, ldb + c4v)));
    const float k3   = fmaf(F.LGTA1, k1, k3b);
    const float k4   = fmaf(-C5, x3, fmaf(-C6, x4, (LB_INV / C4) * c4v));
    k[1] = k1; k[3] = k3; k[4] = k4;
}

// One RK4 step + delay-line update. Returns new X[4] when RECORD.
template <bool RECORD>
__device__ __forceinline__ float zf_step(float X[5], float fd[zf::DELAY], float bd[zf::DELAY],
                                         const Frame& F) {
    using namespace zf;
    const float f_tau = fd[0];
    const float b_tau = bd[0];
    const float Vext  = fmaf(F.TA1, X[1], b_tau);  // stage-1 Vext goes into delay line
    const float nb    = -R_REFL * f_tau;           // new backward wave
    // b_tau_prev == value just written at slot i == nb  (since (step-1-TAU)%9 == step%9)
    const float ldb   = LG_INV * ((b_tau - nb) * INV_DT);  // LG_INV * dbtau, stage-invariant
    const float c4bt  = C4 * b_tau;                        // C4 * b_tau, stage-invariant
    const float C4TA1 = C4 * F.TA1;

    float k1[5], k2[5], k3[5], k4[5], Xt[5];
    // C4*Vext_stage = C4*TA1*x1_stage + C4*b_tau  (stage 1 uses X[1])
    zf_deriv(X, fmaf(C4TA1, X[1], c4bt), F, ldb, k1);
    k1[0] = X[1]; k1[2] = X[3];
#pragma unroll
    for (int j = 0; j < 5; ++j) Xt[j] = fmaf(HALF_DT, k1[j], X[j]);
    zf_deriv(Xt, fmaf(C4TA1, Xt[1], c4bt), F, ldb, k2);
    k2[0] = Xt[1]; k2[2] = Xt[3];
#pragma unroll
    for (int j = 0; j < 5; ++j) Xt[j] = fmaf(HALF_DT, k2[j], X[j]);
    zf_deriv(Xt, fmaf(C4TA1, Xt[1], c4bt), F, ldb, k3);
    k3[0] = Xt[1]; k3[2] = Xt[3];
#pragma unroll
    for (int j = 0; j < 5; ++j) Xt[j] = fmaf(DT, k3[j], X[j]);
    zf_deriv(Xt, fmaf(C4TA1, Xt[1], c4bt), F, ldb, k4);
    k4[0] = Xt[1]; k4[2] = Xt[3];
#pragma unroll
    for (int j = 0; j < 5; ++j)
        X[j] = fmaf(DT6, fmaf(2.0f, k2[j] + k3[j], k1[j] + k4[j]), X[j]);

    // 8-deep FIFO shift (renamed away after 8x unroll of the caller loop)
#pragma unroll
    for (int j = 0; j < DELAY - 1; ++j) { fd[j] = fd[j + 1]; bd[j] = bd[j + 1]; }
    fd[DELAY - 1] = Vext;
    bd[DELAY - 1] = nb;

    return RECORD ? X[4] : 0.0f;
}

// One lane per batch element; 32 lanes per block (one full wave32 per block),
// so each wave runs alone on a SIMD and the serial chain sees no arbitration.
__global__ __launch_bounds__(32) void ZebraFinch_kernel(
        const float* __restrict__ alpha, const float* __restrict__ beta,
        const float* __restrict__ gain, float* __restrict__ out,
        const int* __restrict__ n_frames_p, unsigned total_params) {
    using namespace zf;
    const unsigned nf    = (unsigned)*n_frames_p;  // uniform scalar load
    const unsigned batch = total_params / nf;      // unsigned div: short SALU sequence
    const unsigned b     = blockIdx.x * 32u + threadIdx.x;
    if (b >= batch) return;

    float X[5];
#pragma unroll
    for (int j = 0; j < 5; ++j) X[j] = 1e-11f;
    float fd[DELAY], bd[DELAY];
#pragma unroll
    for (int j = 0; j < DELAY; ++j) { fd[j] = 0.0f; bd[j] = 0.0f; }

    const float* aP = alpha + (size_t)b * nf;
    const float* bP = beta  + (size_t)b * nf;
    const float* gP = gain  + (size_t)b * nf;
    float*       oP = out   + (size_t)b * (size_t)nf * (LSTEPS / SP);  // 256 outputs/frame

    for (unsigned fr = 0; fr < nf; ++fr) {
        const float A1 = gP[fr] * FACTOR;
        Frame F;
        F.ga    = GG * aP[fr];
        F.b     = bP[fr];
        F.TA1   = T_IN * A1;
        F.LGTA1 = LG_INV * T_IN * A1;

        // gfx1250 global_prefetch for next frame's params (clamped, branchless)
        const unsigned nxt = (fr + 1 < nf) ? fr + 1 : fr;
        __builtin_prefetch(aP + nxt, 0, 3);
        __builtin_prefetch(bP + nxt, 0, 3);
        __builtin_prefetch(gP + nxt, 0, 3);

        for (int s = 0; s < LSTEPS; s += 8) {
            float4 o;
            zf_step<false>(X, fd, bd, F);
            o.x = zf_step<true>(X, fd, bd, F);   // odd global steps are recorded
            zf_step<false>(X, fd, bd, F);
            o.y = zf_step<true>(X, fd, bd, F);
            zf_step<false>(X, fd, bd, F);
            o.z = zf_step<true>(X, fd, bd, F);
            zf_step<false>(X, fd, bd, F);
            o.w = zf_step<true>(X, fd, bd, F);
            *reinterpret_cast<float4*>(oP) = o;  // global_store_b128, 16B aligned
            oP += 4;
        }
    }
}

extern "C" void kernel_launch(void* const* d_in, const int* in_sizes, int n_in,
                              void* d_out, int out_size, void* d_ws, size_t ws_size,
                              hipStream_t stream) {
    const float* alpha = (const float*)d_in[0];
    const float* beta  = (const float*)d_in[1];
    const float* gain  = (const float*)d_in[2];
    const int*   nfp   = (const int*)d_in[3];
    float*       out   = (float*)d_out;

    const unsigned total_params = (unsigned)in_sizes[0];      // B * n_frames
    const unsigned blocks       = (total_params + 31u) / 32u; // covers B; surplus blocks exit fast
    ZebraFinch_kernel<<<blocks, 32, 0, stream>>>(alpha, beta, gain, out, nfp, total_params);
}